// OLPHead_14671608283634
// MI455X (gfx1250) — compile-verified
//
#include <hip/hip_runtime.h>
#include <stdint.h>

typedef float v2f __attribute__((ext_vector_type(2)));
typedef float v8f __attribute__((ext_vector_type(8)));

#define N_    2048
#define P_    196
#define D_    768
#define O_    512
#define KSEL  16
#define GIMG  4          // images per block in the local-GEMM kernel
#define APAD  (D_ + 4)   // padded A-row stride (floats): conflict-free LDS

// ---- workspace layout (bytes) ----
// Bp_patch : float2 [(D_/4)*2][O_]   @ 0         (1572864 B)  packed W_patch^T frags
// Bp_fus   : float2 [(2O_/4)*2][O_]  @ 1572864   (2097152 B)  packed W_fusion^T frags
// local    : float  [N_][O_]         @ 3670016   (4194304 B)
#define WS_BPP    0
#define WS_BPF    1572864
#define WS_LOCAL  3670016

// ------------------------------------------------------------------
// Kernel 0: pack weights into per-lane WMMA B fragments.
// Fragment for k-step d0, lane-half h holds {W[o][d0+2h], W[o][d0+2h+1]}
// at Bp[((d0>>2)*2 + h)*O_ + o]  ->  one coalesced b64 load per fragment.
// ------------------------------------------------------------------
__global__ void olp_pack_kernel(const float* __restrict__ Wp,
                                const float* __restrict__ Wf,
                                float2* __restrict__ Bpp,
                                float2* __restrict__ Bpf) {
    int i = blockIdx.x * blockDim.x + threadIdx.x;
    if (i < (D_ / 4) * 2 * O_) {
        int g2 = i / O_, o = i % O_;
        int d = (g2 >> 1) * 4 + (g2 & 1) * 2;
        Bpp[i] = make_float2(Wp[o * D_ + d], Wp[o * D_ + d + 1]);
    }
    if (i < (2 * O_ / 4) * 2 * O_) {
        int g2 = i / O_, o = i % O_;
        int c = (g2 >> 1) * 4 + (g2 & 1) * 2;
        Bpf[i] = make_float2(Wf[o * (2 * O_) + c], Wf[o * (2 * O_) + c + 1]);
    }
}

// ------------------------------------------------------------------
// Kernel 1 (fused): norms -> top-16 -> async gather to LDS -> fp32 WMMA
// GEMM (GIMG x [16x768 @ 768x512]) -> max over 16 -> local.
// One block = GIMG images, 256 threads (8 waves).
// Dynamic LDS:  As[GIMG][16][APAD] | snorm[GIMG][P_] | ridx[GIMG*16]
// ------------------------------------------------------------------
__global__ void __launch_bounds__(256)
olp_local_kernel(const float* __restrict__ pt,
                 const float2* __restrict__ Bpp,
                 float* __restrict__ localbuf) {
    extern __shared__ char smem[];
    float* As    = (float*)smem;                                    // GIMG*16*APAD
    float* snorm = (float*)(smem + GIMG * 16 * APAD * 4);           // GIMG*P_
    int*   ridx  = (int*)  (smem + GIMG * 16 * APAD * 4 + GIMG * P_ * 4);

    const int n0   = blockIdx.x * GIMG;
    const int tid  = threadIdx.x;
    const int lane = tid & 31;
    const int wave = tid >> 5;

    // ---- phase 1: squared L2 norms of all GIMG*P_ patches (float4 loads) ----
    for (int p = wave; p < GIMG * P_; p += 8) {
        const int img = p / P_, pp = p % P_;
        const float4* row =
            (const float4*)(pt + ((size_t)(n0 + img) * P_ + pp) * D_);
        float s = 0.f;
        for (int d4 = lane; d4 < D_ / 4; d4 += 32) {
            float4 v = row[d4];
            s += v.x * v.x + v.y * v.y + v.z * v.z + v.w * v.w;
        }
        #pragma unroll
        for (int m = 16; m >= 1; m >>= 1) s += __shfl_xor(s, m, 32);
        if (lane == 0) snorm[p] = s;
    }
    __syncthreads();

    // ---- phase 2: top-16 per image; wave g selects for image g ----
    if (wave < GIMG) {
        float* sn = snorm + wave * P_;
        for (int k = 0; k < KSEL; ++k) {
            float bv = -1.f; int bi = 0x7fffffff;
            for (int p = lane; p < P_; p += 32) {
                float v = sn[p];
                if (v > bv) { bv = v; bi = p; }
            }
            #pragma unroll
            for (int m = 16; m >= 1; m >>= 1) {
                float ov = __shfl_xor(bv, m, 32);
                int   oi = __shfl_xor(bi, m, 32);
                if (ov > bv || (ov == bv && oi < bi)) { bv = ov; bi = oi; }
            }
            if (lane == 0) { ridx[wave * KSEL + k] = bi; sn[bi] = -1.f; }
        }
    }
    __syncthreads();

    // ---- phase 3: async gather of selected rows into LDS (b128/lane) ----
    // GIMG*16 rows x 192 float4 = 12288 chunks / 256 threads = 48 each.
    for (int i = tid; i < GIMG * KSEL * (D_ / 4); i += 256) {
        const int r12 = i / (D_ / 4);         // 0..GIMG*16-1
        const int c4  = i % (D_ / 4);
        const int img = r12 >> 4;
        const float* gsrc = pt + ((size_t)(n0 + img) * P_ + ridx[r12]) * D_ + c4 * 4;
        const unsigned ldsb = (unsigned)(uintptr_t)(As + r12 * APAD + c4 * 4);
        asm volatile("global_load_async_to_lds_b128 %0, %1, off"
                     :: "v"(ldsb), "v"((const void*)gsrc) : "memory");
    }
    asm volatile("s_wait_asynccnt 0x0" ::: "memory");
    __syncthreads();

    // ---- phase 4: WMMA GEMM, B fragments shared across GIMG images ----
    const int row  = lane & 15;            // A: M / B: N index
    const int h    = lane >> 4;            // lane half -> K{0,1} or K{2,3}
    const int koff = h << 1;
    const int ob   = wave * 64;            // 4 contiguous 16-wide o-tiles per wave

    v8f acc[GIMG][4] = {};
    for (int d0 = 0; d0 < D_; d0 += 4) {
        const float2* bp = Bpp + (size_t)((d0 >> 2) * 2 + h) * O_ + ob + row;
        v2f b[4];
        #pragma unroll
        for (int t = 0; t < 4; ++t) {
            float2 bb = bp[t * 16];
            b[t].x = bb.x; b[t].y = bb.y;
        }
        #pragma unroll
        for (int g = 0; g < GIMG; ++g) {
            v2f a = *(const v2f*)&As[(g * 16 + row) * APAD + d0 + koff];
            #pragma unroll
            for (int t = 0; t < 4; ++t) {
                acc[g][t] = __builtin_amdgcn_wmma_f32_16x16x4_f32(
                    false, a, false, b[t], (short)0, acc[g][t], false, false);
            }
        }
    }

    // ---- phase 5: max over the 16 selected patches (M dim) ----
    #pragma unroll
    for (int g = 0; g < GIMG; ++g) {
        #pragma unroll
        for (int t = 0; t < 4; ++t) {
            float m = acc[g][t][0];
            #pragma unroll
            for (int i = 1; i < 8; ++i) m = fmaxf(m, acc[g][t][i]);
            m = fmaxf(m, __shfl_xor(m, 16, 32));
            if (lane < 16)
                localbuf[(size_t)(n0 + g) * O_ + ob + t * 16 + lane] = m;
        }
    }
}

// ------------------------------------------------------------------
// Kernel 2: fusion GEMM.  16 images per block: X(16x1024) @ Wf^T(1024x512).
// K in two 512-wide LDS phases (cls half, local half); packed B frags.
// Writes raw fused values to d_out.
// ------------------------------------------------------------------
__global__ void __launch_bounds__(256)
olp_fused_kernel(const float* __restrict__ cls,
                 const float* __restrict__ localbuf,
                 const float2* __restrict__ Bpf,
                 float* __restrict__ out) {
    __shared__ float Xs[16][O_ + 4];       // 33 KB

    const int nb   = blockIdx.x * 16;
    const int tid  = threadIdx.x;
    const int lane = tid & 31;
    const int wave = tid >> 5;
    const int row  = lane & 15;
    const int h    = lane >> 4;
    const int koff = h << 1;
    const int ob   = wave * 64;

    v8f acc[4] = {};
    for (int half = 0; half < 2; ++half) {
        const float* src = (half == 0) ? cls : localbuf;
        __syncthreads();
        for (int i = tid; i < 16 * (O_ / 4); i += 256) {
            int r  = i >> 7;
            int c4 = (i & 127) * 4;
            *(float4*)&Xs[r][c4] =
                *(const float4*)(src + (size_t)(nb + r) * O_ + c4);
        }
        __syncthreads();

        const int dg = half * O_;
        for (int d0 = 0; d0 < O_; d0 += 4) {
            v2f a = *(const v2f*)&Xs[row][d0 + koff];
            const float2* bp =
                Bpf + (size_t)(((dg + d0) >> 2) * 2 + h) * O_ + ob + row;
            #pragma unroll
            for (int t = 0; t < 4; ++t) {
                float2 bb = bp[t * 16];
                v2f b; b.x = bb.x; b.y = bb.y;
                acc[t] = __builtin_amdgcn_wmma_f32_16x16x4_f32(
                    false, a, false, b, (short)0, acc[t], false, false);
            }
        }
    }

    // store raw fused: VGPR i holds row M = i + 8*h, col = lane&15
    const int mrow = h * 8;
    #pragma unroll
    for (int t = 0; t < 4; ++t) {
        #pragma unroll
        for (int i = 0; i < 8; ++i) {
            out[(size_t)(nb + mrow + i) * O_ + ob + t * 16 + row] = acc[t][i];
        }
    }
}

// ------------------------------------------------------------------
// Kernel 3: row-wise L2 normalize (clamp 1e-12), in place on d_out.
// ------------------------------------------------------------------
__global__ void __launch_bounds__(256)
olp_norm_kernel(float* __restrict__ out) {
    const int n    = blockIdx.x * 8 + (threadIdx.x >> 5);
    const int lane = threadIdx.x & 31;
    float* rowp = out + (size_t)n * O_;
    float s = 0.f;
    for (int c = lane; c < O_; c += 32) { float v = rowp[c]; s += v * v; }
    #pragma unroll
    for (int m = 16; m >= 1; m >>= 1) s += __shfl_xor(s, m, 32);
    const float inv = 1.0f / fmaxf(sqrtf(s), 1e-12f);
    for (int c = lane; c < O_; c += 32) rowp[c] *= inv;
}

// ------------------------------------------------------------------
extern "C" void kernel_launch(void* const* d_in, const int* in_sizes, int n_in,
                              void* d_out, int out_size, void* d_ws, size_t ws_size,
                              hipStream_t stream) {
    const float* cls = (const float*)d_in[0];   // (N, O)
    const float* pt  = (const float*)d_in[1];   // (N, P, D)
    const float* Wp  = (const float*)d_in[2];   // (O, D)
    const float* Wf  = (const float*)d_in[3];   // (O, 2O)
    float* out = (float*)d_out;                 // (N, O)

    char* ws = (char*)d_ws;
    float2* Bpp      = (float2*)(ws + WS_BPP);
    float2* Bpf      = (float2*)(ws + WS_BPF);
    float*  localbuf = (float*) (ws + WS_LOCAL);

    const int packElems = (2 * O_ / 4) * 2 * O_;           // 262144 (covers both)
    olp_pack_kernel<<<(packElems + 255) / 256, 256, 0, stream>>>(Wp, Wf, Bpp, Bpf);

    const size_t ldsBytes = (size_t)GIMG * 16 * APAD * 4   // A tiles   (197632)
                          + (size_t)GIMG * P_ * 4          // snorm     (3136)
                          + (size_t)GIMG * KSEL * 4;       // ridx      (256)
    olp_local_kernel<<<N_ / GIMG, 256, ldsBytes, stream>>>(pt, Bpp, localbuf);

    olp_fused_kernel<<<N_ / 16, 256, 0, stream>>>(cls, localbuf, Bpf, out);
    olp_norm_kernel<<<N_ / 8, 256, 0, stream>>>(out);
}